// GeoFNO_88287347736733
// MI455X (gfx1250) — compile-verified
//
#include <hip/hip_runtime.h>
#include <math.h>

// GeoFNO on MI455X (gfx1250), fp32 end-to-end using V_WMMA_F32_16X16X4_F32.
// Memory-bound problem (~1.5 TFLOP vs ~100us HBM floor) -> keep fp32 precision,
// optimize data movement: bases_c/s (128MB) are L2-resident, h stored [B][N][C].

#define BB 8
#define NN 16384
#define KK 128
#define CC 128
#define LL 4
#define FCD 128
#define NSPLIT 32

#define NT 32     // n-chunk for forward projection
#define NTP 34    // padded LDS stride
#define KC 32     // contraction chunk for inverse/head
#define KCP 34
#define BP 258    // padded pair-interleaved B stride (2*128 + 2)

typedef float v2f __attribute__((ext_vector_type(2)));
typedef float v8f __attribute__((ext_vector_type(8)));

__device__ __forceinline__ v8f wmma_f32(v2f a, v2f b, v8f c) {
    return __builtin_amdgcn_wmma_f32_16x16x4_f32(false, a, false, b, (short)0, c,
                                                 false, false);
}

__device__ __forceinline__ float gelu_exact(float v) {
    return 0.5f * v * (1.0f + erff(v * 0.7071067811865476f));
}

// ---------------------------------------------------------------- bases
// bases_c[b,n,k] = cos(nodes[b,n,:]·modes[k,:]) * mask ; bases_s likewise.
__global__ __launch_bounds__(128)
void bases_kernel(const float* __restrict__ nodes, const float* __restrict__ nmask,
                  const float* __restrict__ modes,
                  float* __restrict__ bc, float* __restrict__ bs) {
    const int k = threadIdx.x;                 // 0..127
    const int grp = blockIdx.x, b = blockIdx.y;
    const float mk0 = modes[k * 2 + 0];
    const float mk1 = modes[k * 2 + 1];
    for (int r = 0; r < 16; ++r) {
        const long n = (long)b * NN + grp * 16 + r;
        const float t = nodes[n * 2 + 0] * mk0 + nodes[n * 2 + 1] * mk1;
        const float msk = nmask[n];
        float sv, cv;
        sincosf(t, &sv, &cv);
        bc[n * KK + k] = cv * msk;
        bs[n * KK + k] = sv * msk;
    }
}

// ---------------------------------------------------------------- fc0 lift
// h[b,n,c] = x[b,n,:]·fc0_w[:,c] + fc0_b[c]   (IN_DIM = 3)
__global__ __launch_bounds__(128)
void fc0_kernel(const float* __restrict__ x, const float* __restrict__ w,
                const float* __restrict__ bias, float* __restrict__ h) {
    const int c = threadIdx.x;
    const int grp = blockIdx.x, b = blockIdx.y;
    const float w0 = w[c], w1 = w[CC + c], w2 = w[2 * CC + c], bb = bias[c];
    for (int r = 0; r < 16; ++r) {
        const long n = (long)b * NN + grp * 16 + r;
        const float v = fmaf(x[n * 3 + 0], w0,
                        fmaf(x[n * 3 + 1], w1,
                        fmaf(x[n * 3 + 2], w2, bb)));
        h[n * CC + c] = v;
    }
}

// ---------------------------------------------------------------- forward proj
// partial[c,k] += sum_n (h[n,c]*nw[n]) * bases[n,k] over this block's N slice.
// grid: (cblk=2, NSPLIT, B); 256 threads = 8 waves; each wave: 1 c-tile x 4 k-tiles.
__global__ __launch_bounds__(256)
void fwd_proj(const float* __restrict__ h, const float* __restrict__ bc,
              const float* __restrict__ bs, const float* __restrict__ nw,
              float* __restrict__ xcp, float* __restrict__ xsp) {
    __shared__ alignas(16) float lds_hw[64 * NTP];   // [c][n] transposed
    __shared__ alignas(16) float lds_bc[128 * NTP];  // [k][n] transposed
    __shared__ alignas(16) float lds_bs[128 * NTP];
    const int cblk = blockIdx.x, split = blockIdx.y, b = blockIdx.z;
    const int tid = threadIdx.x;
    const int wave = tid >> 5, lane = tid & 31;
    const int ct = wave & 3, khalf = wave >> 2;
    const int m = lane & 15;
    const int rlo = (lane < 16) ? 0 : 2;
    v8f accC[4] = {};
    v8f accS[4] = {};
    const long hb = (long)b * NN;
    const int nbase = split * (NN / NSPLIT);
    for (int n0 = nbase; n0 < nbase + NN / NSPLIT; n0 += NT) {
        __syncthreads();
        // stage h * node_weights, transposed to [c][n]
        for (int j = 0; j < 8; ++j) {
            const int idx = tid + j * 256;        // 32n x 64c
            const int r = idx >> 6, ccol = idx & 63;
            const float w = nw[hb + n0 + r];
            lds_hw[ccol * NTP + r] = h[(hb + n0 + r) * CC + cblk * 64 + ccol] * w;
        }
        // stage bases transposed to [k][n]
        for (int j = 0; j < 16; ++j) {
            const int idx = tid + j * 256;        // 32n x 128k
            const int r = idx >> 7, k = idx & 127;
            lds_bc[k * NTP + r] = bc[(hb + n0 + r) * KK + k];
            lds_bs[k * NTP + r] = bs[(hb + n0 + r) * KK + k];
        }
        __syncthreads();
        for (int n4 = 0; n4 < NT; n4 += 4) {
            const v2f a = *(const v2f*)&lds_hw[(ct * 16 + m) * NTP + n4 + rlo];
#pragma unroll
            for (int j = 0; j < 4; ++j) {
                const int kk = khalf * 64 + j * 16 + m;
                const v2f bcv = *(const v2f*)&lds_bc[kk * NTP + n4 + rlo];
                const v2f bsv = *(const v2f*)&lds_bs[kk * NTP + n4 + rlo];
                accC[j] = wmma_f32(a, bcv, accC[j]);
                accS[j] = wmma_f32(a, bsv, accS[j]);
            }
        }
    }
    // write per-split partials (deterministic; reduced later in fixed order)
    const long pb = (((long)split * BB + b) * CC) * KK;
#pragma unroll
    for (int j = 0; j < 4; ++j) {
        const int k = khalf * 64 + j * 16 + m;
#pragma unroll
        for (int r = 0; r < 8; ++r) {
            const int c = cblk * 64 + ct * 16 + r + ((lane < 16) ? 0 : 8);
            xcp[pb + (long)c * KK + k] = accC[j][r];
            xsp[pb + (long)c * KK + k] = accS[j][r];
        }
    }
}

// ---------------------------------------------------------------- x0_hat
__global__ __launch_bounds__(256)
void x0_reduce(const float* __restrict__ h, const float* __restrict__ nw,
               const float* __restrict__ nmask, float* __restrict__ x0) {
    __shared__ float red[256];
    const int c = blockIdx.x, b = blockIdx.y, tid = threadIdx.x;
    float s = 0.f;
    for (int n = tid; n < NN; n += 256) {
        const long ni = (long)b * NN + n;
        s = fmaf(h[ni * CC + c], nw[ni] * nmask[ni], s);
    }
    red[tid] = s;
    __syncthreads();
    for (int st = 128; st > 0; st >>= 1) {
        if (tid < st) red[tid] += red[tid + st];
        __syncthreads();
    }
    if (tid == 0) x0[b * CC + c] = red[0];
}

// ---------------------------------------------------------------- split reduce
// x_c_hat = sum_s partials ; x_s_hat = -sum_s partials (reference sign)
__global__ __launch_bounds__(256)
void reduce_parts(const float* __restrict__ xcp, const float* __restrict__ xsp,
                  float* __restrict__ xc, float* __restrict__ xs) {
    const long idx = (long)blockIdx.x * 256 + threadIdx.x;
    const long stride = (long)BB * CC * KK;
    if (idx < stride) {
        float sc = 0.f, ss = 0.f;
        for (int s = 0; s < NSPLIT; ++s) {
            sc += xcp[(long)s * stride + idx];
            ss += xsp[(long)s * stride + idx];
        }
        xc[idx] = sc;
        xs[idx] = -ss;
    }
}

// ---------------------------------------------------------------- mode mixing
// f_c = x_c·wc - x_s·ws ; f_s = x_s·wc + x_c·ws ; f_0 = x_0·w0   (contraction i)
__global__ __launch_bounds__(128)
void mix_kernel(const float* __restrict__ xc, const float* __restrict__ xs,
                const float* __restrict__ x0, const float* __restrict__ wc,
                const float* __restrict__ wsm, const float* __restrict__ w0,
                int layer, float* __restrict__ fc, float* __restrict__ fs,
                float* __restrict__ f0) {
    const int o = blockIdx.x, b = blockIdx.y, k = threadIdx.x;
    const long wb = (long)layer * CC * CC;
    float ac = 0.f, as = 0.f;
    for (int i = 0; i < CC; ++i) {
        const float xcv = xc[((long)b * CC + i) * KK + k];
        const float xsv = xs[((long)b * CC + i) * KK + k];
        const float wcv = wc[(wb + (long)i * CC + o) * KK + k];
        const float wsv = wsm[(wb + (long)i * CC + o) * KK + k];
        ac = fmaf(xcv, wcv, ac);
        ac = fmaf(-xsv, wsv, ac);
        as = fmaf(xsv, wcv, as);
        as = fmaf(xcv, wsv, as);
    }
    fc[((long)b * CC + o) * KK + k] = ac;
    fs[((long)b * CC + o) * KK + k] = as;
    if (k == 0) {
        float a0 = 0.f;
        for (int i = 0; i < CC; ++i)
            a0 = fmaf(x0[b * CC + i], w0[wb + (long)i * CC + o], a0);
        f0[b * CC + o] = a0;
    }
}

// ---------------------------------------------------------------- inverse + conv
// h_out[n,o] = f0[o]*mask[n] + 2*f_c·bases_c^T - 2*f_s·bases_s^T + conv_w·h + conv_b
// fused as one 384-long contraction chain into the same WMMA accumulators.
// grid: (N/64, B); 256 threads = 8 waves; each wave: 1 n-tile x 4 o-tiles.
__global__ __launch_bounds__(256)
void inv_conv(const float* __restrict__ bc, const float* __restrict__ bs,
              const float* __restrict__ h, const float* __restrict__ fch,
              const float* __restrict__ fsh, const float* __restrict__ f0h,
              const float* __restrict__ convw, const float* __restrict__ convb,
              const float* __restrict__ nmask, int layer, int do_gelu,
              float* __restrict__ hout) {
    __shared__ alignas(16) float lds_A[64 * KCP];        // [n][k] natural
    __shared__ alignas(16) float lds_B[(KC / 2) * BP];   // pair-interleaved [k][o]
    const int nblk = blockIdx.x, b = blockIdx.y;
    const int tid = threadIdx.x, wave = tid >> 5, lane = tid & 31;
    const int nt = wave & 3, ohalf = wave >> 2;
    const int m = lane & 15;
    const int rl = (lane < 16) ? 0 : 2;
    v8f acc[4] = {};
    const long rowb = (long)b * NN + (long)nblk * 64;
    for (int seg = 0; seg < 3; ++seg) {
        const float* Asrc = (seg == 0) ? bc : ((seg == 1) ? bs : h);
        for (int kc0 = 0; kc0 < 128; kc0 += KC) {
            __syncthreads();
            for (int j = 0; j < 8; ++j) {          // A: 64n x 32k
                const int idx = tid + j * 256;
                const int r = idx >> 5, c = idx & 31;
                lds_A[r * KCP + c] = Asrc[(rowb + r) * 128 + kc0 + c];
            }
            for (int j = 0; j < 16; ++j) {         // B: 32k x 128o
                const int idx = tid + j * 256;
                const int o = idx >> 5, r = idx & 31;
                float v;
                if (seg == 2)
                    v = convw[((long)layer * CC + o) * CC + kc0 + r];
                else if (seg == 0)
                    v = 2.0f * fch[((long)b * CC + o) * KK + kc0 + r];
                else
                    v = -2.0f * fsh[((long)b * CC + o) * KK + kc0 + r];
                lds_B[(r >> 1) * BP + o * 2 + (r & 1)] = v;
            }
            __syncthreads();
            for (int k4 = 0; k4 < KC; k4 += 4) {
                const v2f a = *(const v2f*)&lds_A[(nt * 16 + m) * KCP + k4 + rl];
#pragma unroll
                for (int j = 0; j < 4; ++j) {
                    const int o = ohalf * 64 + j * 16 + m;
                    const v2f bv = *(const v2f*)&lds_B[((k4 + rl) >> 1) * BP + o * 2];
                    acc[j] = wmma_f32(a, bv, acc[j]);
                }
            }
        }
    }
    const float* cb = convb + layer * CC;
#pragma unroll
    for (int j = 0; j < 4; ++j) {
        const int o = ohalf * 64 + j * 16 + m;
        const float f0v = f0h[b * CC + o];
        const float cbv = cb[o];
#pragma unroll
        for (int r = 0; r < 8; ++r) {
            const long n = rowb + nt * 16 + r + ((lane < 16) ? 0 : 8);
            float v = acc[j][r] + f0v * nmask[n] + cbv;
            if (do_gelu) v = gelu_exact(v);
            hout[n * CC + o] = v;
        }
    }
}

// ---------------------------------------------------------------- fc1+gelu+fc2
__global__ __launch_bounds__(256)
void head_kernel(const float* __restrict__ h, const float* __restrict__ fc1w,
                 const float* __restrict__ fc1b, const float* __restrict__ fc2w,
                 const float* __restrict__ fc2b, const float* __restrict__ nmask,
                 float* __restrict__ out) {
    __shared__ alignas(16) float lds_A[64 * KCP];
    __shared__ alignas(16) float lds_B[(KC / 2) * BP];
    __shared__ alignas(16) float lds_t[64 * 130];
    const int nblk = blockIdx.x, b = blockIdx.y;
    const int tid = threadIdx.x, wave = tid >> 5, lane = tid & 31;
    const int nt = wave & 3, fhalf = wave >> 2;
    const int m = lane & 15;
    const int rl = (lane < 16) ? 0 : 2;
    v8f acc[4] = {};
    const long rowb = (long)b * NN + (long)nblk * 64;
    for (int kc0 = 0; kc0 < CC; kc0 += KC) {
        __syncthreads();
        for (int j = 0; j < 8; ++j) {
            const int idx = tid + j * 256;
            const int r = idx >> 5, c = idx & 31;
            lds_A[r * KCP + c] = h[(rowb + r) * CC + kc0 + c];
        }
        for (int j = 0; j < 16; ++j) {
            const int idx = tid + j * 256;
            const int r = idx >> 7, f = idx & 127;
            lds_B[(r >> 1) * BP + f * 2 + (r & 1)] = fc1w[(long)(kc0 + r) * FCD + f];
        }
        __syncthreads();
        for (int k4 = 0; k4 < KC; k4 += 4) {
            const v2f a = *(const v2f*)&lds_A[(nt * 16 + m) * KCP + k4 + rl];
#pragma unroll
            for (int j = 0; j < 4; ++j) {
                const int f = fhalf * 64 + j * 16 + m;
                const v2f bv = *(const v2f*)&lds_B[((k4 + rl) >> 1) * BP + f * 2];
                acc[j] = wmma_f32(a, bv, acc[j]);
            }
        }
    }
    __syncthreads();
#pragma unroll
    for (int j = 0; j < 4; ++j) {
        const int f = fhalf * 64 + j * 16 + m;
        const float bf = fc1b[f];
#pragma unroll
        for (int r = 0; r < 8; ++r) {
            const int nloc = nt * 16 + r + ((lane < 16) ? 0 : 8);
            lds_t[nloc * 130 + f] = gelu_exact(acc[j][r] + bf);
        }
    }
    __syncthreads();
    if (tid < 64) {
        float s = 0.f;
        for (int f = 0; f < FCD; ++f) s = fmaf(lds_t[tid * 130 + f], fc2w[f], s);
        const long n = rowb + tid;
        out[n] = (s + fc2b[0]) * nmask[n];
    }
}

// ---------------------------------------------------------------- launcher
extern "C" void kernel_launch(void* const* d_in, const int* in_sizes, int n_in,
                              void* d_out, int out_size, void* d_ws, size_t ws_size,
                              hipStream_t stream) {
    const float* x       = (const float*)d_in[0];
    const float* nodes   = (const float*)d_in[1];
    const float* nmask   = (const float*)d_in[2];
    const float* nweight = (const float*)d_in[3];
    const float* modes   = (const float*)d_in[4];
    const float* fc0w    = (const float*)d_in[5];
    const float* fc0b    = (const float*)d_in[6];
    const float* wc      = (const float*)d_in[7];
    const float* wsm     = (const float*)d_in[8];
    const float* w0      = (const float*)d_in[9];
    const float* convw   = (const float*)d_in[10];
    const float* convb   = (const float*)d_in[11];
    const float* fc1w    = (const float*)d_in[12];
    const float* fc1b    = (const float*)d_in[13];
    const float* fc2w    = (const float*)d_in[14];
    const float* fc2b    = (const float*)d_in[15];
    float* out = (float*)d_out;

    const size_t BNK = (size_t)BB * NN * KK;   // 16.78M floats
    const size_t BNC = (size_t)BB * NN * CC;
    const size_t BCK = (size_t)BB * CC * KK;   // 131072
    float* basesC = (float*)d_ws;
    float* basesS = basesC + BNK;
    float* hA     = basesS + BNK;
    float* hB     = hA + BNC;
    float* xcp    = hB + BNC;
    float* xsp    = xcp + (size_t)NSPLIT * BCK;
    float* xch    = xsp + (size_t)NSPLIT * BCK;
    float* xsh    = xch + BCK;
    float* x0h    = xsh + BCK;
    float* fch    = x0h + (size_t)BB * CC;
    float* fsh    = fch + BCK;
    float* f0h    = fsh + BCK;

    bases_kernel<<<dim3(NN / 16, BB), 128, 0, stream>>>(nodes, nmask, modes,
                                                        basesC, basesS);
    fc0_kernel<<<dim3(NN / 16, BB), 128, 0, stream>>>(x, fc0w, fc0b, hA);

    float* hcur = hA;
    float* hnxt = hB;
    for (int l = 0; l < LL; ++l) {
        fwd_proj<<<dim3(2, NSPLIT, BB), 256, 0, stream>>>(hcur, basesC, basesS,
                                                          nweight, xcp, xsp);
        x0_reduce<<<dim3(CC, BB), 256, 0, stream>>>(hcur, nweight, nmask, x0h);
        reduce_parts<<<dim3((int)(BCK / 256)), 256, 0, stream>>>(xcp, xsp, xch, xsh);
        mix_kernel<<<dim3(CC, BB), 128, 0, stream>>>(xch, xsh, x0h, wc, wsm, w0,
                                                     l, fch, fsh, f0h);
        inv_conv<<<dim3(NN / 64, BB), 256, 0, stream>>>(basesC, basesS, hcur, fch,
                                                        fsh, f0h, convw, convb,
                                                        nmask, l, (l < LL - 1) ? 1 : 0,
                                                        hnxt);
        float* t = hcur; hcur = hnxt; hnxt = t;
    }
    head_kernel<<<dim3(NN / 64, BB), 256, 0, stream>>>(hcur, fc1w, fc1b, fc2w,
                                                       fc2b, nmask, out);
}